// CrossAttentionFusion_69973607186976
// MI455X (gfx1250) — compile-verified
//
#include <hip/hip_runtime.h>

typedef __attribute__((ext_vector_type(2))) float v2f;
typedef __attribute__((ext_vector_type(8))) float v8f;

#define SLEN 4096
#define NPIX 16384
#define HDIM 4

// D = A(16x4,f32) * B(4x16,f32) + C(16x16,f32)  -- exact f32 WMMA, K=4
static __device__ __forceinline__ v8f wmma4(v2f a, v2f b, v8f c) {
  return __builtin_amdgcn_wmma_f32_16x16x4_f32(false, a, false, b, (short)0, c,
                                               false, false);
}

static __device__ __forceinline__ void ln16(const float* __restrict__ x,
                                            const float* __restrict__ g,
                                            const float* __restrict__ be,
                                            float* __restrict__ y) {
  float m = 0.f;
#pragma unroll
  for (int i = 0; i < 16; ++i) m += x[i];
  m *= 0.0625f;
  float v = 0.f;
#pragma unroll
  for (int i = 0; i < 16; ++i) { float d = x[i] - m; v += d * d; }
  v *= 0.0625f;
  float r = rsqrtf(v + 1e-5f);
#pragma unroll
  for (int i = 0; i < 16; ++i) y[i] = (x[i] - m) * r * g[i] + be[i];
}

// ---------------- Kernel 1: per-pixel proj + LN + QKV ----------------
__global__ void prep_kernel(const float* __restrict__ sample,
                            const float* __restrict__ w_rgb, const float* __restrict__ b_rgb,
                            const float* __restrict__ g_rgb, const float* __restrict__ be_rgb,
                            const float* __restrict__ w_dep, const float* __restrict__ b_dep,
                            const float* __restrict__ g_dep, const float* __restrict__ be_dep,
                            const float* __restrict__ in_w, const float* __restrict__ in_b,
                            float* __restrict__ rgbseq, float* __restrict__ qws,
                            float* __restrict__ kws, float* __restrict__ vws) {
  int p = blockIdx.x * blockDim.x + threadIdx.x;
  if (p >= NPIX) return;
  int b = p >> 12, s = p & 4095;
  const float* sb = sample + (size_t)(b * 12) * SLEN + s;
  float rgb[4], dep[4];
#pragma unroll
  for (int c = 0; c < 4; ++c) { rgb[c] = sb[c * SLEN]; dep[c] = sb[(8 + c) * SLEN]; }
  float rs[16], ds[16], rq[16], dkv[16];
#pragma unroll
  for (int e = 0; e < 16; ++e) {
    float ar = b_rgb[e], ad = b_dep[e];
#pragma unroll
    for (int c = 0; c < 4; ++c) {
      ar += w_rgb[e * 4 + c] * rgb[c];
      ad += w_dep[e * 4 + c] * dep[c];
    }
    rs[e] = ar; ds[e] = ad;
    rgbseq[(size_t)p * 16 + e] = ar;
  }
  ln16(rs, g_rgb, be_rgb, rq);
  ln16(ds, g_dep, be_dep, dkv);
#pragma unroll
  for (int e = 0; e < 16; ++e) {
    float qv = in_b[e], kv = in_b[16 + e], vv = in_b[32 + e];
#pragma unroll
    for (int f = 0; f < 16; ++f) {
      qv += dkv[f] * in_w[e * 16 + f];           // q from depth (LN)
      kv += rq[f] * in_w[(16 + e) * 16 + f];     // k from rgb (LN)
      vv += rq[f] * in_w[(32 + e) * 16 + f];     // v from rgb (LN)
    }
    int h = e >> 2, d = e & 3;
    size_t idx = (size_t)(b * 4 + h) * (SLEN * 4) + (size_t)s * 4 + d;
    qws[idx] = qv; kws[idx] = kv; vws[idx] = vv;
  }
}

// ---------------- Kernel 2: flash attention, 1 wave per 16-query tile ----------------
// 64 keys per softmax round; softmax denominator folded into WMMA via a
// ones-column (column 4) appended to the zero-padded V operand.
__global__ void __launch_bounds__(32) attn_kernel(const float* __restrict__ qws,
                                                  const float* __restrict__ kws,
                                                  const float* __restrict__ vws,
                                                  float* __restrict__ ctxws) {
  __shared__ float pbuf[16 * 64];  // 16 query rows x 64 key probs
  int lane = threadIdx.x;
  int qt = blockIdx.x;   // 0..255 query tile
  int bh = blockIdx.y;   // 0..15  (b*NH + h)
  const float* Q = qws + (size_t)bh * (SLEN * 4);
  const float* K = kws + (size_t)bh * (SLEN * 4);
  const float* V = vws + (size_t)bh * (SLEN * 4);
  int n = lane & 15;
  int half = lane >> 4;
  int r0 = half * 2;
  // A fragment = 16 queries x 4 dims, pre-scaled by 1/sqrt(HD) = 0.5
  v2f aq;
  aq.x = Q[(size_t)(qt * 16 + n) * 4 + r0] * 0.5f;
  aq.y = Q[(size_t)(qt * 16 + n) * 4 + r0 + 1] * 0.5f;
  float mrow[8];
#pragma unroll
  for (int i = 0; i < 8; ++i) mrow[i] = -3.0e38f;
  v8f acc0 = {}, acc1 = {};  // ctx cols 0..3, running denom in col 4
  // preload K fragments for slab 0
  v2f bk[4];
#pragma unroll
  for (int j = 0; j < 4; ++j) {
    bk[j].x = K[(size_t)(j * 16 + n) * 4 + r0];
    bk[j].y = K[(size_t)(j * 16 + n) * 4 + r0 + 1];
  }
  for (int it = 0; it < SLEN / 64; ++it) {
    int base = it * 64;
    // 4 independent score WMMAs (64 keys)
    v8f z = {};
    v8f sc[4];
#pragma unroll
    for (int j = 0; j < 4; ++j) sc[j] = wmma4(aq, bk[j], z);
    // software pipeline: fetch next slab's K frags, prefetch V
    if (it + 1 < SLEN / 64) {
      int nb = base + 64;
#pragma unroll
      for (int j = 0; j < 4; ++j) {
        bk[j].x = K[(size_t)(nb + j * 16 + n) * 4 + r0];
        bk[j].y = K[(size_t)(nb + j * 16 + n) * 4 + r0 + 1];
      }
      __builtin_prefetch(&V[(size_t)nb * 4], 0, 3);
      __builtin_prefetch(&V[(size_t)(nb + 32) * 4], 0, 3);
    }
    // one online-softmax max update per 64 keys
#pragma unroll
    for (int i = 0; i < 8; ++i) {
      float mx = fmaxf(fmaxf(sc[0][i], sc[1][i]), fmaxf(sc[2][i], sc[3][i]));
#pragma unroll
      for (int off = 1; off < 16; off <<= 1) mx = fmaxf(mx, __shfl_xor(mx, off, 32));
      float mnew = fmaxf(mrow[i], mx);
      float rescale = __expf(mrow[i] - mnew);
      mrow[i] = mnew;
      acc0[i] = acc0[i] * rescale;  // rescales ctx columns AND denom column
      acc1[i] = acc1[i] * rescale;
      int row = i + 8 * half;
#pragma unroll
      for (int j = 0; j < 4; ++j)
        pbuf[row * 64 + j * 16 + n] = __expf(sc[j][i] - mnew);
    }
    __syncthreads();
    // acc += P(16x64) @ [V | 1 | 0...](64x16): 16 K=4 chunks on 2 accumulators
#pragma unroll
    for (int c = 0; c < 16; ++c) {
      v2f ap;
      ap.x = pbuf[n * 64 + c * 4 + r0];
      ap.y = pbuf[n * 64 + c * 4 + r0 + 1];
      int key = base + c * 4 + r0;
      v2f bv;
      bv.x = (n < HDIM) ? V[(size_t)key * 4 + n] : (n == HDIM ? 1.f : 0.f);
      bv.y = (n < HDIM) ? V[(size_t)(key + 1) * 4 + n] : (n == HDIM ? 1.f : 0.f);
      if (c & 1)
        acc1 = wmma4(ap, bv, acc1);
      else
        acc0 = wmma4(ap, bv, acc0);
    }
    __syncthreads();
  }
  int b = bh >> 2, h = bh & 3;
  int srcl = (lane & 16) + 4;  // lane holding the denom column of this half
#pragma unroll
  for (int i = 0; i < 8; ++i) {
    float num = acc0[i] + acc1[i];
    float l = __shfl(num, srcl, 32);
    if (n < HDIM) {
      int sidx = qt * 16 + i + 8 * half;
      ctxws[(size_t)(b * SLEN + sidx) * 16 + h * 4 + n] = num / l;
    }
  }
}

// ---------------- Kernel 3: out-proj + residual + LN + MLP + fuse + concat ----------------
__global__ void post_kernel(const float* __restrict__ sample,
                            const float* __restrict__ ctxws, const float* __restrict__ rgbseq,
                            const float* __restrict__ ow, const float* __restrict__ ob,
                            const float* __restrict__ g2, const float* __restrict__ be2,
                            const float* __restrict__ f1w, const float* __restrict__ f1b,
                            const float* __restrict__ f2w, const float* __restrict__ f2b,
                            const float* __restrict__ wf, const float* __restrict__ bf,
                            float* __restrict__ xcat) {
  int p = blockIdx.x * blockDim.x + threadIdx.x;
  if (p >= NPIX) return;
  int b = p >> 12, s = p & 4095;
  float ctx[16], xf[16], l2[16], h1[16], y[16];
#pragma unroll
  for (int e = 0; e < 16; ++e) ctx[e] = ctxws[(size_t)p * 16 + e];
#pragma unroll
  for (int e = 0; e < 16; ++e) {
    float a = ob[e];
#pragma unroll
    for (int f = 0; f < 16; ++f) a += ctx[f] * ow[e * 16 + f];
    xf[e] = a + rgbseq[(size_t)p * 16 + e];
  }
  ln16(xf, g2, be2, l2);
#pragma unroll
  for (int e = 0; e < 16; ++e) {
    float a = f1b[e];
#pragma unroll
    for (int f = 0; f < 16; ++f) a += l2[f] * f1w[e * 16 + f];
    h1[e] = 0.5f * a * (1.f + erff(a * 0.70710678118654752f));  // exact GELU
  }
#pragma unroll
  for (int e = 0; e < 16; ++e) {
    float a = f2b[e];
#pragma unroll
    for (int f = 0; f < 16; ++f) a += h1[f] * f2w[e * 16 + f];
    y[e] = xf[e] + a;
  }
  const float* sb = sample + (size_t)(b * 12) * SLEN + s;
  float* xb = xcat + (size_t)(b * 16) * SLEN + s;
#pragma unroll
  for (int c = 0; c < 4; ++c) {
    float fu = bf[c];
#pragma unroll
    for (int e = 0; e < 16; ++e) fu += y[e] * wf[c * 16 + e];
    xb[c * SLEN] = sb[c * SLEN];               // rgb
    xb[(4 + c) * SLEN] = sb[(4 + c) * SLEN];   // flow
    xb[(8 + c) * SLEN] = fu;                   // fused
    xb[(12 + c) * SLEN] = sb[(8 + c) * SLEN];  // depth
  }
}

// ---------------- Kernel 4: 3x3 conv 16->320 as implicit-GEMM WMMA ----------------
__global__ void __launch_bounds__(32) conv_kernel(const float* __restrict__ xcat,
                                                  const float* __restrict__ cw,
                                                  const float* __restrict__ cb,
                                                  float* __restrict__ out) {
  __shared__ float wl[2304];      // [tap][ic][oc] = 9*16*16
  __shared__ float xin[16 * 54];  // [ic][3 rows][18 cols] haloed input tile
  int lane = threadIdx.x;
  int tile = blockIdx.x;      // 0..255 pixel tile (row y0, 16-wide x strip)
  int oc0 = blockIdx.y * 16;  // out-channel tile
  int b = blockIdx.z;
  int y0 = tile >> 2;
  int x0 = (tile & 3) * 16;
  for (int idx = lane; idx < 2304; idx += 32) {
    int oc = idx & 15, ic = (idx >> 4) & 15, tap = idx >> 8;
    wl[idx] = cw[(size_t)(oc0 + oc) * 144 + ic * 9 + tap];
  }
  for (int idx = lane; idx < 16 * 54; idx += 32) {
    int ic = idx / 54, rem = idx % 54;
    int dy = rem / 18, xx = rem % 18;
    int gy = y0 - 1 + dy, gx = x0 - 1 + xx;
    float v = 0.f;
    if (gy >= 0 && gy < 64 && gx >= 0 && gx < 64)
      v = xcat[(size_t)(b * 16 + ic) * SLEN + gy * 64 + gx];
    xin[idx] = v;
  }
  __syncthreads();
  int n = lane & 15;    // A-frag row (pixel) == B-frag col (oc)
  int half = lane >> 4;
  int r0 = half * 2;
  v8f acc = {};
#pragma unroll
  for (int tap = 0; tap < 9; ++tap) {
    int dy = tap / 3, dx = tap % 3;
#pragma unroll
    for (int ch = 0; ch < 4; ++ch) {
      int ic = ch * 4 + r0;
      v2f a;  // A = 16 pixels x 4 in-channels at this tap
      a.x = xin[ic * 54 + dy * 18 + n + dx];
      a.y = xin[(ic + 1) * 54 + dy * 18 + n + dx];
      v2f w;  // B = 4 in-channels x 16 out-channels
      w.x = wl[tap * 256 + ic * 16 + n];
      w.y = wl[tap * 256 + (ic + 1) * 16 + n];
      acc = wmma4(a, w, acc);  // 36 WMMAs per output tile
    }
  }
  float bias = cb[oc0 + n];
#pragma unroll
  for (int i = 0; i < 8; ++i) {
    int px = i + 8 * half;
    out[(size_t)(b * 320 + oc0 + n) * SLEN + y0 * 64 + x0 + px] = acc[i] + bias;
  }
}

extern "C" void kernel_launch(void* const* d_in, const int* in_sizes, int n_in,
                              void* d_out, int out_size, void* d_ws, size_t ws_size,
                              hipStream_t stream) {
  (void)in_sizes; (void)n_in; (void)out_size; (void)ws_size;
  const float* sample = (const float*)d_in[0];
  const float* w_proj_rgb = (const float*)d_in[1];
  const float* b_proj_rgb = (const float*)d_in[2];
  const float* g_norm_rgb = (const float*)d_in[3];
  const float* be_norm_rgb = (const float*)d_in[4];
  const float* w_proj_depth = (const float*)d_in[5];
  const float* b_proj_depth = (const float*)d_in[6];
  const float* g_norm_depth = (const float*)d_in[7];
  const float* be_norm_depth = (const float*)d_in[8];
  const float* in_proj_w = (const float*)d_in[9];
  const float* in_proj_b = (const float*)d_in[10];
  const float* out_proj_w = (const float*)d_in[11];
  const float* out_proj_b = (const float*)d_in[12];
  const float* fc1_w = (const float*)d_in[13];
  const float* fc1_b = (const float*)d_in[14];
  const float* fc2_w = (const float*)d_in[15];
  const float* fc2_b = (const float*)d_in[16];
  const float* g_norm2 = (const float*)d_in[17];
  const float* be_norm2 = (const float*)d_in[18];
  const float* w_fuse = (const float*)d_in[19];
  const float* b_fuse = (const float*)d_in[20];
  const float* conv_w = (const float*)d_in[21];
  const float* conv_b = (const float*)d_in[22];

  float* ws = (float*)d_ws;
  const size_t CH = (size_t)NPIX * 16;  // 262144 floats per buffer
  float* rgbseq = ws;
  float* qws = ws + CH;
  float* kws = ws + 2 * CH;
  float* vws = ws + 3 * CH;
  float* ctxws = ws + 4 * CH;
  float* xcat = ws + 5 * CH;

  prep_kernel<<<dim3(NPIX / 256), dim3(256), 0, stream>>>(
      sample, w_proj_rgb, b_proj_rgb, g_norm_rgb, be_norm_rgb, w_proj_depth,
      b_proj_depth, g_norm_depth, be_norm_depth, in_proj_w, in_proj_b, rgbseq,
      qws, kws, vws);

  attn_kernel<<<dim3(SLEN / 16, 16), dim3(32), 0, stream>>>(qws, kws, vws, ctxws);

  post_kernel<<<dim3(NPIX / 256), dim3(256), 0, stream>>>(
      sample, ctxws, rgbseq, out_proj_w, out_proj_b, g_norm2, be_norm2, fc1_w,
      fc1_b, fc2_w, fc2_b, w_fuse, b_fuse, xcat);

  conv_kernel<<<dim3(256, 20, 4), dim3(32), 0, stream>>>(xcat, conv_w, conv_b,
                                                         (float*)d_out);
}